// GaussMixtureCombinedW_74620761800827
// MI455X (gfx1250) — compile-verified
//
#include <hip/hip_runtime.h>
#include <hip/hip_bf16.h>

// ---------------------------------------------------------------------------
// GaussMixtureCombinedW for MI455X (gfx1250, wave32, WMMA)
// N=524288 rows, F=128 features, C=32 classes, EPS=10
// Core: quad = ft^2 @ inv2v.T + ft @ (-mean/var).T + Kc   via V_WMMA_F32_16X16X4_F32
// Persistent blocks: 1024 blocks x 4 row-tiles (128 rows each) to amortize
// LDS staging of the class tensors and reduce atomics to 1 per block.
// ---------------------------------------------------------------------------

#define GN 524288
#define GF 128
#define GC 32
#define GEPS 10.0f
#define TILES_PER_BLOCK 4
#define ROWS_PER_TILE 128

typedef __attribute__((ext_vector_type(2))) float v2f;
typedef __attribute__((ext_vector_type(8))) float v8f;

static __device__ __forceinline__ v8f wmma_f32(v2f a, v2f b, v8f c) {
  // 8 args: (neg_a, A, neg_b, B, c_mod, C, reuse_a, reuse_b)
  return __builtin_amdgcn_wmma_f32_16x16x4_f32(
      false, a, false, b, (short)0, c, false, false);
}

// ---------------------------------------------------------------------------
// Prep 1: per-class tensors.  grid = C blocks, block = F threads.
//   inv2v[c,f] = 1/(2*(sigma+EPS)^2)
//   nmv[c,f]   = -mean/(sigma+EPS)^2
//   Kc[c]      = sum_f mean^2 * inv2v
//   S[c]       = sum_f log(sigma+EPS)
// ---------------------------------------------------------------------------
__global__ __launch_bounds__(GF) void prep_class_kernel(
    const float* __restrict__ mean, const float* __restrict__ sigma,
    float* __restrict__ inv2v, float* __restrict__ nmv,
    float* __restrict__ Kc, float* __restrict__ S) {
  const int c = blockIdx.x;
  const int f = threadIdx.x;
  const int i = c * GF + f;

  const float sg  = sigma[i] + GEPS;
  const float var = sg * sg;
  const float iv  = 1.0f / (2.0f * var);
  const float m   = mean[i];
  inv2v[i] = iv;
  nmv[i]   = -m / var;

  __shared__ float red[GF];
  // Kc reduction
  red[f] = m * m * iv;
  __syncthreads();
  for (int s = GF / 2; s > 0; s >>= 1) {
    if (f < s) red[f] += red[f + s];
    __syncthreads();
  }
  if (f == 0) Kc[c] = red[0];
  __syncthreads();
  // S reduction
  red[f] = __logf(sg);
  __syncthreads();
  for (int s = GF / 2; s > 0; s >>= 1) {
    if (f < s) red[f] += red[f + s];
    __syncthreads();
  }
  if (f == 0) S[c] = red[0];
}

// ---------------------------------------------------------------------------
// Prep 2: W[c,o] = exp(S_c - S_o) * p_o / p_c  (0 on diagonal); zero d_out.
// 1 block of C*C = 1024 threads.
// ---------------------------------------------------------------------------
__global__ __launch_bounds__(GC * GC) void prep_w_kernel(
    const float* __restrict__ S, const float* __restrict__ p,
    float* __restrict__ W, float* __restrict__ out) {
  const int t = threadIdx.x;
  const int c = t >> 5;
  const int o = t & 31;
  W[t] = (c == o) ? 0.0f : __expf(S[c] - S[o]) * p[o] / p[c];
  if (t == 0) out[0] = 0.0f;
}

// ---------------------------------------------------------------------------
// Main fused kernel: 256 threads (8 waves), persistent over 4 tiles of 128
// rows; 1024 blocks total.
// ---------------------------------------------------------------------------
__global__ __launch_bounds__(256) void gmm_main_kernel(
    const float* __restrict__ ft, const int* __restrict__ y,
    const float* __restrict__ inv2v, const float* __restrict__ nmv,
    const float* __restrict__ Kc, const float* __restrict__ W,
    float* __restrict__ out) {
  // LDS: B matrices padded to 130 floats/row (bank-conflict-free b64 frags),
  //      per-wave q tiles padded to 33, W copy, Kc copy, reduction buffer.
  __shared__ float sBinv[GC * 130];
  __shared__ float sBnmv[GC * 130];
  __shared__ float sQ[8][16 * 33];
  __shared__ float sW[GC * GC];
  __shared__ float sKc[GC];
  __shared__ float sRed[128];

  const int tid = threadIdx.x;

  // Stage class tensors once per block (coalesced).
  for (int i = tid; i < GC * GF; i += 256) {
    const int c = i >> 7;
    const int f = i & 127;
    sBinv[c * 130 + f] = inv2v[i];
    sBnmv[c * 130 + f] = nmv[i];
  }
  for (int i = tid; i < GC * GC; i += 256) sW[i] = W[i];
  if (tid < GC) sKc[tid] = Kc[tid];
  __syncthreads();

  const int wave = tid >> 5;      // 0..7
  const int lane = tid & 31;
  const int lh   = lane >> 4;     // lane half: 0 -> K{0,1}, 1 -> K{2,3}
  const int lm   = lane & 15;

  // B fragment sources: column class = lm (+16 for tile 1), K base 2*lh.
  const float* bi0 = &sBinv[lm * 130 + 2 * lh];
  const float* bi1 = &sBinv[(lm + 16) * 130 + 2 * lh];
  const float* bm0 = &sBnmv[lm * 130 + 2 * lh];
  const float* bm1 = &sBnmv[(lm + 16) * 130 + 2 * lh];

  const float kc0 = sKc[lm];
  const float kc1 = sKc[lm + 16];

  float accContrib = 0.0f;  // per-thread sum of log1p(out_i)/N over tiles

  const int tileBase = blockIdx.x * TILES_PER_BLOCK;

  for (int t = 0; t < TILES_PER_BLOCK; ++t) {
    const int rowTile = tileBase + t;
    const int rowBase = rowTile * ROWS_PER_TILE + wave * 16;

    // A fragment source: row = rowBase + lm, K base offset 2*lh (v2f = 8B).
    const float* aptr = ft + (size_t)(rowBase + lm) * GF + 2 * lh;

    v8f acc0 = {};  // classes 0..15
    v8f acc1 = {};  // classes 16..31

#pragma unroll 4
    for (int kk = 0; kk < GF; kk += 4) {
      __builtin_prefetch((const void*)(aptr + kk + 64), 0, 1);
      const v2f a  = *(const v2f*)(aptr + kk);
      const v2f a2 = a * a;
      const v2f vbi0 = *(const v2f*)(bi0 + kk);
      const v2f vbi1 = *(const v2f*)(bi1 + kk);
      const v2f vbm0 = *(const v2f*)(bm0 + kk);
      const v2f vbm1 = *(const v2f*)(bm1 + kk);
      acc0 = wmma_f32(a2, vbi0, acc0);   // + ft^2 * inv2v
      acc0 = wmma_f32(a,  vbm0, acc0);   // + ft * (-mean/var)
      acc1 = wmma_f32(a2, vbi1, acc1);
      acc1 = wmma_f32(a,  vbm1, acc1);
    }

    // Spill q = quad + Kc to per-wave LDS tile.  C/D layout: N=lm, M=j+8*lh.
#pragma unroll
    for (int j = 0; j < 8; ++j) {
      const int r = j + 8 * lh;
      sQ[wave][r * 33 + lm]      = acc0[j] + kc0;
      sQ[wave][r * 33 + 16 + lm] = acc1[j] + kc1;
    }
    __syncthreads();

    // Per-row tail: out_i = sum_o W[c_i,o] * exp(q_ci - q_o)
    if (tid < 128) {
      const int w2  = tid >> 4;
      const int r2  = tid & 15;
      const int row = rowTile * ROWS_PER_TILE + tid;
      const int c   = y[row];
      const float* qrow = &sQ[w2][r2 * 33];
      const float  qc   = qrow[c];
      const float* wrow = &sW[c * GC];
      float s = 0.0f;
#pragma unroll
      for (int o = 0; o < GC; ++o) s += wrow[o] * __expf(qc - qrow[o]);
      accContrib += log1pf(s) * (1.0f / (float)GN);
    }
    __syncthreads();  // protect sQ before next tile overwrites it
  }

  // Block reduction of 128 per-thread accumulators, one atomic per block.
  if (tid < 128) sRed[tid] = accContrib;
  __syncthreads();
  for (int s = 64; s > 0; s >>= 1) {
    if (tid < s) sRed[tid] += sRed[tid + s];
    __syncthreads();
  }
  if (tid == 0) atomicAdd(out, sRed[0]);
}

// ---------------------------------------------------------------------------
// Host launcher
// ---------------------------------------------------------------------------
extern "C" void kernel_launch(void* const* d_in, const int* in_sizes, int n_in,
                              void* d_out, int out_size, void* d_ws, size_t ws_size,
                              hipStream_t stream) {
  (void)in_sizes; (void)n_in; (void)out_size; (void)ws_size;

  const float* ft    = (const float*)d_in[0];   // [N, F]
  const float* mean  = (const float*)d_in[1];   // [C, F]
  const float* sigma = (const float*)d_in[2];   // [C, F]
  const float* prob  = (const float*)d_in[3];   // [C]
  const int*   y     = (const int*)d_in[4];     // [N]
  float* out = (float*)d_out;

  // Workspace layout (floats): inv2v[4096] | nmv[4096] | Kc[32] | S[32] | W[1024]
  float* wsf     = (float*)d_ws;
  float* w_inv2v = wsf;
  float* w_nmv   = wsf + GC * GF;
  float* w_Kc    = wsf + 2 * GC * GF;
  float* w_S     = wsf + 2 * GC * GF + GC;
  float* w_W     = wsf + 2 * GC * GF + 2 * GC;

  prep_class_kernel<<<GC, GF, 0, stream>>>(mean, sigma, w_inv2v, w_nmv, w_Kc, w_S);
  prep_w_kernel<<<1, GC * GC, 0, stream>>>(w_S, prob, w_W, out);

  const int nBlocks = GN / (ROWS_PER_TILE * TILES_PER_BLOCK);  // 1024
  gmm_main_kernel<<<nBlocks, 256, 0, stream>>>(ft, y, w_inv2v, w_nmv, w_Kc, w_W, out);
}